// RNNTLoss_77867757077051
// MI455X (gfx1250) — compile-verified
//
#include <hip/hip_runtime.h>

typedef __attribute__((ext_vector_type(2))) float v2f;
typedef __attribute__((ext_vector_type(8))) float v8f;

#define BB   8
#define TT   128
#define UU   64
#define VV   512
#define UP1  65
#define UP1P 66                      // padded stride for blank rows (8B alignment)
#define NROWS (BB * TT * UP1)        // 66560, divisible by 16

__device__ __forceinline__ float lae(float x, float y) {
  float m = fmaxf(x, y);
  if (m == -__builtin_inff()) return -__builtin_inff();
  float d = fabsf(x - y);
  return m + __logf(1.0f + __expf(-d));
}

// ---------------------------------------------------------------------------
// Kernel 1: fused log-softmax + gather.  One wave per 16 rows of 512.
// Pass A: row max (HBM stream).  Pass B: exp on TRANS pipe, row-sum on the
// matrix pipe via V_WMMA_F32_16X16X4_F32 with B = ones (re-read hits cache).
// ---------------------------------------------------------------------------
__global__ __launch_bounds__(256) void rnnt_lse_gather(
    const float* __restrict__ x, const int* __restrict__ y,
    float* __restrict__ blank_lp, float* __restrict__ emit_lp) {
  __shared__ float sums[8 * 16];
  const int lane = threadIdx.x & 31;
  const int wv   = threadIdx.x >> 5;
  const int wid  = blockIdx.x * 8 + wv;
  const int r0   = wid * 16;
  const int m    = lane & 15;          // row within wave tile (A-matrix M)
  const int h    = lane >> 4;          // K half: lanes 0-15 -> K=0,1; 16-31 -> K=2,3
  const float2* rp2 = (const float2*)(x + (size_t)(r0 + m) * VV + 2 * h);

  // ---- pass A: row max ----
  float mx = -__builtin_inff();
#pragma unroll 8
  for (int c = 0; c < 128; ++c) {
    float2 v = rp2[2 * c];
    mx = fmaxf(mx, fmaxf(v.x, v.y));
  }
  mx = fmaxf(mx, __shfl_xor(mx, 16, 32));   // combine the two K-halves of each row

  // ---- pass B: sum of exp via WMMA accumulation ----
  v8f acc0 = {}; v8f acc1 = {};
  v2f ones; ones[0] = 1.0f; ones[1] = 1.0f;
#pragma unroll 4
  for (int c = 0; c < 128; c += 2) {
    float2 v0 = rp2[2 * c];
    float2 v1 = rp2[2 * c + 2];
    v2f a, b;
    a[0] = __expf(v0.x - mx); a[1] = __expf(v0.y - mx);
    b[0] = __expf(v1.x - mx); b[1] = __expf(v1.y - mx);
    acc0 = __builtin_amdgcn_wmma_f32_16x16x4_f32(false, a, false, ones,
                                                 (short)0, acc0, false, false);
    acc1 = __builtin_amdgcn_wmma_f32_16x16x4_f32(false, b, false, ones,
                                                 (short)0, acc1, false, false);
  }
  v8f acc = acc0 + acc1;   // D[m,n] == sumexp(row m) for every column n

  // D layout: VGPR r, lanes 0-15 -> M=r ; lanes 16-31 -> M=r+8.
  if ((lane & 15) == 0) {
#pragma unroll
    for (int r = 0; r < 8; ++r) sums[wv * 16 + h * 8 + r] = acc[r];
  }
  // same-wave LDS ops are in order; compiler waits DScnt before the read below
  if (lane < 16) {
    const int row = r0 + lane;                 // lane == m here
    const float lse = mx + __logf(sums[wv * 16 + lane]);
    const int u  = row % UP1;
    const int bt = row / UP1;
    blank_lp[(size_t)bt * UP1P + u] = x[(size_t)row * VV] - lse;
    if (u < UU) {
      const int b_  = bt / TT;
      const int yix = y[b_ * UU + u];          // in [1, V1)
      emit_lp[(size_t)bt * UU + u] = x[(size_t)row * VV + yix] - lse;
    }
  }
}

// ---------------------------------------------------------------------------
// Kernel 2: alpha DP.  One wave per batch.  The u-recurrence is linearized:
// alpha_t[u] = P[u] + logaddexp-scan(a[k] - P[k])[u],  P = exclusive sum(e).
// Both scans are wave32 shuffle scans (2 u-slots per lane, u=64 tail combine).
// ---------------------------------------------------------------------------
__device__ __forceinline__ float pick_alpha(float alo, float ahi, float a64,
                                            int yl, int lane) {
  float v  = (yl & 1) ? ahi : alo;
  float sh = __shfl(v, yl >> 1, 32);
  return (yl == 64) ? a64 : sh;
}

__global__ __launch_bounds__(256) void rnnt_alpha_dp(
    const float* __restrict__ blank_lp, const float* __restrict__ emit_lp,
    const int* __restrict__ x_lens, const int* __restrict__ y_lens,
    float* __restrict__ out) {
  __shared__ float lossSm[BB];
  const int lane = threadIdx.x & 31;
  const int b    = threadIdx.x >> 5;
  const int xl = x_lens[b];
  const int yl = y_lens[b];
  const float NEG = -__builtin_inff();

  // ---- t = 0 : alpha0[u] = exclusive prefix sum of emit[b,0,:] ----
  float2 ev = *(const float2*)(emit_lp + (size_t)(b * TT) * UU + 2 * lane);
  float pr = ev.x + ev.y;
  float s  = pr;
#pragma unroll
  for (int o = 1; o < 32; o <<= 1) {
    float t2 = __shfl_up(s, o, 32);
    if (lane >= o) s += t2;
  }
  float excl = s - pr;
  float alpha_lo = excl;            // alpha[2l]
  float alpha_hi = excl + ev.x;     // alpha[2l+1]
  float alpha64  = __shfl(s, 31, 32);
  float loss_b = 0.0f;
  if (xl == 1) {
    float av = pick_alpha(alpha_lo, alpha_hi, alpha64, yl, lane);
    loss_b = -(av + blank_lp[(size_t)(b * TT) * UP1P + yl]);
  }

  // ---- t = 1 .. T-1 ----
  for (int t = 1; t < TT; ++t) {
    const float* blrow = blank_lp + (size_t)(b * TT + t - 1) * UP1P;
    float2 bl  = *(const float2*)(blrow + 2 * lane);
    float bl64 = blrow[64];
    float a_lo = alpha_lo + bl.x;
    float a_hi = alpha_hi + bl.y;
    float a64  = alpha64  + bl64;

    float2 ev2 = *(const float2*)(emit_lp + (size_t)(b * TT + t) * UU + 2 * lane);
    // additive exclusive scan -> P
    float pr2 = ev2.x + ev2.y;
    float ss  = pr2;
#pragma unroll
    for (int o = 1; o < 32; o <<= 1) {
      float t2 = __shfl_up(ss, o, 32);
      if (lane >= o) ss += t2;
    }
    float ex  = ss - pr2;
    float Plo = ex, Phi = ex + ev2.x;

    // logaddexp inclusive scan of c = a - P
    float c_lo = a_lo - Plo, c_hi = a_hi - Phi;
    float pc = lae(c_lo, c_hi);
    float w  = pc;
#pragma unroll
    for (int o = 1; o < 32; o <<= 1) {
      float t2 = __shfl_up(w, o, 32);
      if (lane >= o) w = lae(w, t2);
    }
    float exc = __shfl_up(w, 1, 32);
    if (lane == 0) exc = NEG;
    alpha_lo = Plo + lae(exc, c_lo);
    alpha_hi = Phi + w;                       // lae(exc, pc) == inclusive w

    float alpha63 = __shfl(alpha_hi, 31, 32);
    float e63     = __shfl(ev2.y, 31, 32);
    alpha64 = lae(a64, alpha63 + e63);

    if (t == xl - 1) {
      float av = pick_alpha(alpha_lo, alpha_hi, alpha64, yl, lane);
      loss_b = -(av + blank_lp[(size_t)(b * TT + xl - 1) * UP1P + yl]);
    }
  }

  if (lane == 0) lossSm[b] = loss_b;
  __syncthreads();
  if (threadIdx.x == 0) {
    float sm = 0.0f;
#pragma unroll
    for (int i = 0; i < BB; ++i) sm += lossSm[i];
    out[0] = sm / (float)BB;
  }
}

extern "C" void kernel_launch(void* const* d_in, const int* in_sizes, int n_in,
                              void* d_out, int out_size, void* d_ws, size_t ws_size,
                              hipStream_t stream) {
  (void)in_sizes; (void)n_in; (void)out_size; (void)ws_size;
  const float* x   = (const float*)d_in[0];
  const int*   y   = (const int*)d_in[1];
  const int*   xl  = (const int*)d_in[2];
  const int*   yls = (const int*)d_in[3];
  float* ws       = (float*)d_ws;
  float* blank_lp = ws;                         // BB*TT*UP1P floats (padded)
  float* emit_lp  = ws + (size_t)BB * TT * UP1P;

  const int nblk = NROWS / 16 / 8;              // 520 blocks of 8 waves
  rnnt_lse_gather<<<nblk, 256, 0, stream>>>(x, y, blank_lp, emit_lp);
  rnnt_alpha_dp<<<1, 256, 0, stream>>>(blank_lp, emit_lp, xl, yls, (float*)d_out);
}